// AttnDecoderRNN_12232066859333
// MI455X (gfx1250) — compile-verified
//
#include <hip/hip_runtime.h>
#include <hip/hip_bf16.h>
#include <math.h>

#define HID 1024
#define VOC 50257
#define LEN 18

typedef float v2f __attribute__((ext_vector_type(2)));
typedef float v8f __attribute__((ext_vector_type(8)));

// ---------------------------------------------------------------------------
// Kernel 1: embedding gather + additive attention (scores, softmax, apply).
// Writes xcat[0:H] = embedded, xcat[H:2H] = attn_applied, attnw_out[0:L] = w.
// Single block of 256 threads; total traffic ~220 KB -> negligible.
// ---------------------------------------------------------------------------
__global__ void k_attn(const int* __restrict__ ids,
                       const float* __restrict__ hidden,
                       const float* __restrict__ enc,
                       const float* __restrict__ emb,
                       const float* __restrict__ attn_W,
                       const float* __restrict__ attn_b,
                       float* __restrict__ xcat,
                       float* __restrict__ attnw_out) {
  __shared__ float red[8];
  __shared__ float sc[LEN];
  __shared__ float wgt[LEN];
  const int t = threadIdx.x;
  const int id = ids[0];
  const float* e = emb + (size_t)id * HID;

  // embedded -> xcat[0:H]
  for (int j = t; j < HID; j += 256) xcat[j] = e[j];

  // scores over concat(embedded, hidden) . attn_W[l] + attn_b[l]
  for (int l = 0; l < LEN; ++l) {
    const float* wrow = attn_W + (size_t)l * 2 * HID;
    float p = 0.0f;
    for (int j = t; j < HID; j += 256)
      p = fmaf(e[j], wrow[j], fmaf(hidden[j], wrow[HID + j], p));
    for (int o = 16; o; o >>= 1) p += __shfl_xor(p, o, 32);
    if ((t & 31) == 0) red[t >> 5] = p;
    __syncthreads();
    if (t == 0) {
      float s = 0.0f;
      for (int w = 0; w < 8; ++w) s += red[w];
      sc[l] = s + attn_b[l];
    }
    __syncthreads();
  }

  // softmax over L=18 (thread 0; trivial)
  if (t == 0) {
    float m = sc[0];
    for (int l = 1; l < LEN; ++l) m = fmaxf(m, sc[l]);
    float s = 0.0f;
    for (int l = 0; l < LEN; ++l) { float ev = expf(sc[l] - m); wgt[l] = ev; s += ev; }
    float inv = 1.0f / s;
    for (int l = 0; l < LEN; ++l) { wgt[l] *= inv; attnw_out[l] = wgt[l]; }
  }
  __syncthreads();

  // attn_applied = w @ encoder_outputs -> xcat[H:2H]
  for (int j = t; j < HID; j += 256) {
    float acc = 0.0f;
    for (int l = 0; l < LEN; ++l) acc = fmaf(wgt[l], enc[l * HID + j], acc);
    xcat[HID + j] = acc;
  }
}

// ---------------------------------------------------------------------------
// Generic WMMA GEMV: y[n] = act( sum_k x[k] * W[n*K + k] + b[n] )
// One wave computes 16 output rows via V_WMMA_F32_16X16X4_F32:
//   A(16x4)  : x[k0..k0+3] replicated down M  (lane half selects K pair)
//   B(4x16)  : 16 weight rows, each lane streams an 8-byte (2-float) chunk
//   D(16x16) : all M rows identical; row 0 (VGPR0, lanes 0..15) is the result
// Bandwidth-bound by design: 256 B of weights per WMMA issue.
// ---------------------------------------------------------------------------
__global__ void k_wmma_gemv(const float* __restrict__ W,
                            const float* __restrict__ x,
                            const float* __restrict__ b,
                            float* __restrict__ y,
                            int N, int K, int act) {
  const int lane = threadIdx.x & 31;
  const int wave = threadIdx.x >> 5;
  const int nbase = (blockIdx.x * (blockDim.x >> 5) + wave) * 16;
  if (nbase >= N) return;                 // wave-uniform: EXEC stays all-ones

  int row = nbase + (lane & 15);
  if (row >= N) row = N - 1;              // clamp loads; stores are guarded
  const int koff = (lane >> 4) * 2;       // lanes 0-15 -> K 0,1 ; 16-31 -> K 2,3
  const float* wp = W + (size_t)row * K + koff;
  const float* xp = x + koff;

  v8f c = {0.f, 0.f, 0.f, 0.f, 0.f, 0.f, 0.f, 0.f};
  for (int k0 = 0; k0 < K; k0 += 4) {
    v2f a  = *(const v2f*)(xp + k0);      // broadcast vector chunk (L2-hot)
    v2f bm = *(const v2f*)(wp + k0);      // global_load_b64 of weight row
    c = __builtin_amdgcn_wmma_f32_16x16x4_f32(
        /*neg_a=*/false, a, /*neg_b=*/false, bm,
        /*c_mod=*/(short)0, c, /*reuse_a=*/false, /*reuse_b=*/false);
  }

  const int n = nbase + lane;             // D: VGPR0 lanes 0..15 = M=0 row
  if (lane < 16 && n < N) {
    float v = c[0] + b[n];
    if (act == 1) v = fmaxf(v, 0.0f);
    y[n] = v;
  }
}

// ---------------------------------------------------------------------------
// Kernel: GRU gate nonlinearity (PyTorch r,z,n ordering).
// ---------------------------------------------------------------------------
__global__ void k_gru(const float* __restrict__ gi,
                      const float* __restrict__ gh,
                      const float* __restrict__ h,
                      float* __restrict__ hnew_ws,
                      float* __restrict__ hnew_out) {
  const int j = blockIdx.x * blockDim.x + threadIdx.x;
  if (j >= HID) return;
  const float r = 1.0f / (1.0f + expf(-(gi[j] + gh[j])));
  const float z = 1.0f / (1.0f + expf(-(gi[HID + j] + gh[HID + j])));
  const float n = tanhf(gi[2 * HID + j] + r * gh[2 * HID + j]);
  const float v = (1.0f - z) * n + z * h[j];
  hnew_ws[j]  = v;
  hnew_out[j] = v;
}

// ---------------------------------------------------------------------------
// Kernel: in-place log-softmax over d_out[0:V]. Single block, 1024 threads.
// Logits ~200 KB -> resident in L2 after the GEMV; three passes are cheap.
// ---------------------------------------------------------------------------
__global__ void k_logsoftmax(float* __restrict__ out, int V) {
  __shared__ float red[32];
  __shared__ float bc;
  const int t = threadIdx.x;
  const int nt = blockDim.x;

  // pass 1: max
  float m = -INFINITY;
  for (int i = t; i < V; i += nt) m = fmaxf(m, out[i]);
  for (int o = 16; o; o >>= 1) m = fmaxf(m, __shfl_xor(m, o, 32));
  if ((t & 31) == 0) red[t >> 5] = m;
  __syncthreads();
  if (t == 0) {
    float r = -INFINITY;
    for (int w = 0; w < 32; ++w) r = fmaxf(r, red[w]);
    bc = r;
  }
  __syncthreads();
  const float gmax = bc;
  __syncthreads();

  // pass 2: sum of exp
  float s = 0.0f;
  for (int i = t; i < V; i += nt) s += expf(out[i] - gmax);
  for (int o = 16; o; o >>= 1) s += __shfl_xor(s, o, 32);
  if ((t & 31) == 0) red[t >> 5] = s;
  __syncthreads();
  if (t == 0) {
    float r = 0.0f;
    for (int w = 0; w < 32; ++w) r += red[w];
    bc = gmax + logf(r);
  }
  __syncthreads();
  const float lse = bc;

  // pass 3: rewrite in place
  for (int i = t; i < V; i += nt) out[i] = out[i] - lse;
}

// ---------------------------------------------------------------------------
// Host-side launch sequence.
// ---------------------------------------------------------------------------
extern "C" void kernel_launch(void* const* d_in, const int* in_sizes, int n_in,
                              void* d_out, int out_size, void* d_ws, size_t ws_size,
                              hipStream_t stream) {
  const int*   ids      = (const int*)  d_in[0];
  const float* hidden   = (const float*)d_in[1];   // [H]
  const float* enc      = (const float*)d_in[2];   // [L,H]
  const float* emb      = (const float*)d_in[3];   // [V,H]
  const float* attn_W   = (const float*)d_in[4];   // [L,2H]
  const float* attn_b   = (const float*)d_in[5];   // [L]
  const float* comb_W   = (const float*)d_in[6];   // [H,2H]
  const float* comb_b   = (const float*)d_in[7];   // [H]
  const float* gru_Wih  = (const float*)d_in[8];   // [3H,H]
  const float* gru_Whh  = (const float*)d_in[9];   // [3H,H]
  const float* gru_bih  = (const float*)d_in[10];  // [3H]
  const float* gru_bhh  = (const float*)d_in[11];  // [3H]
  const float* out_W    = (const float*)d_in[12];  // [V,H]
  const float* out_b    = (const float*)d_in[13];  // [V]

  float* out = (float*)d_out;          // [V log_probs | H h_new | L attn_w]
  float* ws  = (float*)d_ws;

  // workspace layout (floats)
  float* xcat = ws;                    // 2H  : [embedded | attn_applied]
  float* xv   = ws + 2 * HID;          // H   : relu(comb)
  float* gi   = ws + 3 * HID;          // 3H
  float* gh   = ws + 6 * HID;          // 3H
  float* hn   = ws + 9 * HID;          // H   : h_new

  const int rowsPerBlock = (256 / 32) * 16;   // 8 waves * 16 rows = 128

  // 1) attention
  k_attn<<<1, 256, 0, stream>>>(ids, hidden, enc, emb, attn_W, attn_b,
                                xcat, out + VOC + HID);

  // 2) x = relu(comb_W @ xcat + comb_b)       N=H, K=2H
  k_wmma_gemv<<<(HID + rowsPerBlock - 1) / rowsPerBlock, 256, 0, stream>>>(
      comb_W, xcat, comb_b, xv, HID, 2 * HID, /*act=*/1);

  // 3) gi = gru_Wih @ x + bih                 N=3H, K=H
  k_wmma_gemv<<<(3 * HID + rowsPerBlock - 1) / rowsPerBlock, 256, 0, stream>>>(
      gru_Wih, xv, gru_bih, gi, 3 * HID, HID, 0);

  // 4) gh = gru_Whh @ h + bhh                 N=3H, K=H
  k_wmma_gemv<<<(3 * HID + rowsPerBlock - 1) / rowsPerBlock, 256, 0, stream>>>(
      gru_Whh, hidden, gru_bhh, gh, 3 * HID, HID, 0);

  // 5) GRU gates -> h_new (ws + d_out[V..V+H))
  k_gru<<<(HID + 255) / 256, 256, 0, stream>>>(gi, gh, hidden, hn, out + VOC);

  // 6) logits = out_W @ h_new + out_b -> d_out[0:V)   (206 MB weight stream)
  k_wmma_gemv<<<(VOC + rowsPerBlock - 1) / rowsPerBlock, 256, 0, stream>>>(
      out_W, hn, out_b, out, VOC, HID, 0);

  // 7) in-place log_softmax over d_out[0:V)
  k_logsoftmax<<<1, 1024, 0, stream>>>(out, VOC);
}